// BrixiaCustomLoss_29515015258485
// MI455X (gfx1250) — compile-verified
//
#include <hip/hip_runtime.h>
#include <cstdint>

#define NBLK   4096
#define BLOCK  256

constexpr unsigned B_COUNT = 1048576u;
constexpr unsigned N_AREAS = 6u;
constexpr unsigned PAIRS   = B_COUNT * N_AREAS;        // 6,291,456
constexpr unsigned GSZ     = NBLK * BLOCK;             // 1,048,576
constexpr unsigned ITERS   = PAIRS / GSZ;              // exactly 6
static_assert(ITERS * GSZ == PAIRS, "grid must tile PAIRS exactly");

// Issue a per-lane 16B async copy global -> LDS (CDNA5, ASYNCcnt-tracked).
// GVS addressing: mem_addr = SGPR64 + VGPR32_offset. First operand = LDS byte addr VGPR.
#define ASYNC_LD_B128(ldsoff, voff, base)                                   \
  asm volatile("global_load_async_to_lds_b128 %0, %1, %2"                   \
               :: "v"(ldsoff), "v"(voff), "s"(base) : "memory")

__global__ __launch_bounds__(BLOCK) void brixia_main(
    const float* __restrict__ outs,       // (B,6,4) f32, 16B per (b,a) pair
    const int*   __restrict__ labels,     // (B,6) i32
    float* __restrict__ lab_pred,         // (B,6)
    float* __restrict__ prob_pred,        // (B,6)
    float* __restrict__ regress,          // (B,)  (area == 5 only)
    float* __restrict__ partials)         // NBLK block partial sums
{
  __shared__ float4 sbuf[2][BLOCK];       // double-buffered async staging (8 KB)
  __shared__ float  sred[BLOCK];

  const unsigned tid = threadIdx.x;
  const unsigned i0  = blockIdx.x * BLOCK + tid;

  // Raw LDS byte offsets of this thread's two staging slots
  // (flat LDS aperture: addr[31:0] is the LDS offset).
  const uint32_t lds0 = (uint32_t)(uintptr_t)&sbuf[0][tid];
  const uint32_t lds1 = (uint32_t)(uintptr_t)&sbuf[1][tid];
  const unsigned long long base = (unsigned long long)(uintptr_t)outs;

  // Prologue: stage iterations 0 and 1 (2 async loads in flight).
  ASYNC_LD_B128(lds0, (i0 + 0u * GSZ) * 16u, base);
  ASYNC_LD_B128(lds1, (i0 + 1u * GSZ) * 16u, base);

  // Incremental (batch, area) decomposition of i = i0 + k*GSZ.
  // GSZ = 6*174762 + 4  ->  per step: area += 4 (mod 6), batch += 174762 (+carry)
  unsigned area = i0 % N_AREAS;
  unsigned bidx = i0 / N_AREAS;

  constexpr float LOG2E = 1.4426950408889634f;   // log2(e)
  constexpr float LN2   = 0.6931471805599453f;   // ln(2)

  float partial = 0.0f;

  for (unsigned k = 0; k < ITERS; ++k) {
    const unsigned i = i0 + k * GSZ;

    // Oldest outstanding async load (stage k) must have landed.
    if (k < ITERS - 1) asm volatile("s_wait_asynccnt 1" ::: "memory");
    else               asm volatile("s_wait_asynccnt 0" ::: "memory");

    const float4 x4 = sbuf[k & 1u][tid];                  // ds_load_b128
    const int    lab = __builtin_nontemporal_load(&labels[i]);

    // Ensure the LDS read retired before overwriting the slot (WAR), then
    // kick off the stage-(k+2) copy into it.
    asm volatile("s_wait_dscnt 0" ::: "memory");
    if (k + 2u < ITERS)
      ASYNC_LD_B128((k & 1u) ? lds1 : lds0, (i0 + (k + 2u) * GSZ) * 16u, base);

    const float x0 = x4.x, x1 = x4.y, x2 = x4.z, x3 = x4.w;
    // One scaling feeds every exponential: exp(z) = 2^(z*log2e).
    const float y0 = x0 * LOG2E, y1 = x1 * LOG2E;
    const float y2 = x2 * LOG2E, y3 = x3 * LOG2E;

    // ---- BCE with logits: max(x,0) - x*onehot + log1p(exp(-|x|)) ----
    float bce = fmaxf(x0, 0.f) + fmaxf(x1, 0.f) + fmaxf(x2, 0.f) + fmaxf(x3, 0.f);
    // Branchless 2-level bit-mux on lab (0..3): pure v_cndmask, no EXEC churn.
    const bool b0 = (lab & 1) != 0, b1 = (lab & 2) != 0;
    const float sel01 = b0 ? x1 : x0;
    const float sel23 = b0 ? x3 : x2;
    bce -= b1 ? sel23 : sel01;
    // t_c = exp(-|x_c|) = 2^(-|y_c|);  sum log1p(t_c) = ln2 * log2(prod(1+t_c))
    const float t0 = __builtin_amdgcn_exp2f(-fabsf(y0));
    const float t1 = __builtin_amdgcn_exp2f(-fabsf(y1));
    const float t2 = __builtin_amdgcn_exp2f(-fabsf(y2));
    const float t3 = __builtin_amdgcn_exp2f(-fabsf(y3));
    const float prod = (1.0f + t0) * (1.0f + t1) * (1.0f + t2) * (1.0f + t3);
    bce += LN2 * __builtin_amdgcn_logf(prod);

    // ---- Softmax stats: e_c = 2^(y_c - ymax) ----
    const float ym = fmaxf(fmaxf(y0, y1), fmaxf(y2, y3));
    const float e0 = __builtin_amdgcn_exp2f(y0 - ym);
    const float e1 = __builtin_amdgcn_exp2f(y1 - ym);
    const float e2 = __builtin_amdgcn_exp2f(y2 - ym);
    const float e3 = __builtin_amdgcn_exp2f(y3 - ym);
    const float s    = e0 + e1 + e2 + e3;
    const float invs = __builtin_amdgcn_rcpf(s);
    const float pmax = invs;              // 2^(ymax-ymax) == 1 exactly

    // argmax with first-index tie rule (jnp.argmax): select float directly.
    float bestf = 0.0f; float by = y0;
    if (y1 > by) { by = y1; bestf = 1.0f; }
    if (y2 > by) { by = y2; bestf = 2.0f; }
    if (y3 > by) { by = y3; bestf = 3.0f; }

    const float regressed = (e1 + 2.0f * e2 + 3.0f * e3) * invs;

    partial += 0.25f * bce + fabsf(regressed - (float)lab);

    // ---- Streaming (write-once) outputs: keep them out of L2 ----
    __builtin_nontemporal_store(bestf, &lab_pred[i]);
    __builtin_nontemporal_store(pmax,  &prob_pred[i]);
    if (area == N_AREAS - 1u)
      __builtin_nontemporal_store(regressed, &regress[bidx]);

    // advance (batch, area) by GSZ = 6*174762 + 4
    area += 4u; bidx += 174762u;
    if (area >= N_AREAS) { area -= N_AREAS; bidx += 1u; }
  }

  // Deterministic per-block reduction.
  sred[tid] = partial;
  __syncthreads();
  #pragma unroll
  for (unsigned off = BLOCK / 2; off > 0; off >>= 1) {
    if (tid < off) sred[tid] += sred[tid + off];
    __syncthreads();
  }
  if (tid == 0) partials[blockIdx.x] = sred[0];
}

__global__ __launch_bounds__(BLOCK) void brixia_reduce(
    const float* __restrict__ partials, float* __restrict__ total)
{
  __shared__ float sred[BLOCK];
  float acc = 0.0f;
  for (unsigned i = threadIdx.x; i < NBLK; i += BLOCK)
    acc += partials[i];
  sred[threadIdx.x] = acc;
  __syncthreads();
  #pragma unroll
  for (unsigned off = BLOCK / 2; off > 0; off >>= 1) {
    if (threadIdx.x < off) sred[threadIdx.x] += sred[threadIdx.x + off];
    __syncthreads();
  }
  if (threadIdx.x == 0) {
    // total = ALPHA*sum_bce/(B*4) + (1-ALPHA)*sum_reg/B, ALPHA = 0.5;
    // partial already folded the /4 into the bce term.
    total[0] = sred[0] * (0.5f / (float)B_COUNT);
  }
}

extern "C" void kernel_launch(void* const* d_in, const int* in_sizes, int n_in,
                              void* d_out, int out_size, void* d_ws, size_t ws_size,
                              hipStream_t stream) {
  (void)in_sizes; (void)n_in; (void)out_size; (void)ws_size;
  const float* outs   = (const float*)d_in[0];
  const int*   labels = (const int*)d_in[1];
  float* o        = (float*)d_out;
  float* partials = (float*)d_ws;                 // 4096 floats = 16 KB

  brixia_main<<<NBLK, BLOCK, 0, stream>>>(
      outs, labels,
      o + 1,                    // labels_predicted (B*6)
      o + 1 + PAIRS,            // probs_predicted  (B*6)
      o + 1 + 2u * PAIRS,       // regression_output (B)
      partials);
  brixia_reduce<<<1, BLOCK, 0, stream>>>(partials, o);  // o[0] = total_loss
}